// Recommender_55018531062595
// MI455X (gfx1250) — compile-verified
//
#include <hip/hip_runtime.h>

typedef __attribute__((ext_vector_type(2))) float v2f;
typedef __attribute__((ext_vector_type(8))) float v8f;

#define N_ENT   200000
#define N_USERS 100000
#define CH      64
#define N_EDGES 1000000
#define NNZ_    1000000
#define NRELM1  25
#define NF      4
#define SCORE_SCALE 0.08838834764831845f  // 1/(2*sqrt(32))

// workspace layout (in floats)
#define OFF_V    0                        // v_tab[25][64]
#define OFF_RR   1600                     // ||weight[r]||^2, 25
#define OFF_DIS  1632                     // disen[4][64]
#define OFF_MAX1 1920                     // per-entity softmax max (ordered-u32), N_ENT
#define OFF_SUM1 (OFF_MAX1 + N_ENT)
#define OFF_MAX2 (OFF_SUM1 + N_ENT)
#define OFF_SUM2 (OFF_MAX2 + N_ENT)
#define OFF_CNT  (OFF_SUM2 + N_ENT)
#define OFF_SR   (OFF_CNT + N_ENT)        // per-edge score_r / exp, E
#define OFF_ST   (OFF_SR + N_EDGES)       // per-edge score_trip / exp, E

__device__ __forceinline__ unsigned f2ord(float f) {
  unsigned u = __float_as_uint(f);
  return (u & 0x80000000u) ? ~u : (u | 0x80000000u);
}
__device__ __forceinline__ float ord2f(unsigned o) {
  return (o & 0x80000000u) ? __uint_as_float(o & 0x7fffffffu) : __uint_as_float(~o);
}

__global__ void k_zero(float* __restrict__ p, long n) {
  long i = (long)blockIdx.x * blockDim.x + threadIdx.x;
  long s = (long)gridDim.x * blockDim.x;
  for (; i < n; i += s) p[i] = 0.0f;
}

// tiny precompute: v_tab = W @ (W^T @ weight^T), rel norms, disen = softmax(att) @ weight
__global__ void k_pre(const float* __restrict__ weight, const float* __restrict__ datt,
                      const float* __restrict__ W, float* __restrict__ ws) {
  __shared__ float sW[64 * 64];
  __shared__ float skA[NRELM1 * 64];
  __shared__ float sDA[NF * NRELM1];
  int tid = threadIdx.x;
  for (int i = tid; i < 64 * 64; i += blockDim.x) sW[i] = W[i];
  __syncthreads();
  for (int idx = tid; idx < NRELM1 * 64; idx += blockDim.x) {
    int r = idx >> 6, j = idx & 63;
    float s = 0.0f;
    for (int m = 0; m < 64; ++m) s += weight[r * 64 + m] * sW[m * 64 + j];
    skA[idx] = s;  // k_rel[r][j] = (weight[r] @ W)[j]
  }
  if (tid < NRELM1) {
    float s = 0.0f;
    for (int m = 0; m < 64; ++m) { float w = weight[tid * 64 + m]; s += w * w; }
    ws[OFF_RR + tid] = s;
  }
  if (tid < NF) {
    float mx = -1e30f;
    for (int r = 0; r < NRELM1; ++r) mx = fmaxf(mx, datt[tid * NRELM1 + r]);
    float d = 0.0f;
    for (int r = 0; r < NRELM1; ++r) {
      float e = __expf(datt[tid * NRELM1 + r] - mx);
      sDA[tid * NRELM1 + r] = e; d += e;
    }
    float inv = 1.0f / d;
    for (int r = 0; r < NRELM1; ++r) sDA[tid * NRELM1 + r] *= inv;
  }
  __syncthreads();
  for (int idx = tid; idx < NRELM1 * 64; idx += blockDim.x) {
    int r = idx >> 6, i = idx & 63;
    float s = 0.0f;
    for (int j = 0; j < 64; ++j) s += sW[i * 64 + j] * skA[r * 64 + j];
    ws[OFF_V + idx] = s;  // v_tab[r][i] = (W @ k_rel[r])[i]
  }
  for (int idx = tid; idx < NF * 64; idx += blockDim.x) {
    int f = idx >> 6, c = idx & 63;
    float s = 0.0f;
    for (int r = 0; r < NRELM1; ++r) s += sDA[f * NRELM1 + r] * weight[r * 64 + c];
    ws[OFF_DIS + idx] = s;
  }
}

// pass 1: score_r[e] = dot(ent[head], v_tab[rel]) * scale ; segment max
__global__ void k_edge1(const float* __restrict__ ent, const int* __restrict__ eidx,
                        const int* __restrict__ etype, const float* __restrict__ vtab,
                        float* __restrict__ score_r, unsigned* __restrict__ max1) {
  int e = blockIdx.x * blockDim.x + threadIdx.x;
  if (e >= N_EDGES) return;
  int h = eidx[e];
  int t = eidx[N_EDGES + e];
  int r = etype[e] - 1;
  __builtin_prefetch(ent + (long)t * CH, 0, 0);  // tail row used in pass 3 -> global_prefetch_b8
  const float4* hr = (const float4*)(ent + (long)h * CH);
  const float4* vr = (const float4*)(vtab + r * CH);
  float s = 0.0f;
#pragma unroll
  for (int i = 0; i < 16; ++i) {
    float4 a = hr[i], b = vr[i];
    s += a.x * b.x + a.y * b.y + a.z * b.z + a.w * b.w;
  }
  s *= SCORE_SCALE;
  score_r[e] = s;
  atomicMax(max1 + h, f2ord(s));
}

// pass 2/4: exp(score - max), segment sum  (in-place on score buffer)
__global__ void k_exp_sum(const int* __restrict__ eidx, float* __restrict__ sc,
                          const unsigned* __restrict__ mx, float* __restrict__ sm) {
  int e = blockIdx.x * blockDim.x + threadIdx.x;
  if (e >= N_EDGES) return;
  int h = eidx[e];
  float ex = __expf(sc[e] - ord2f(mx[h]));
  sc[e] = ex;
  atomicAdd(sm + h, ex);
}

// pass 3: rel_attn = exp/sum ; score_trip = h.t + rel_attn^2 * ||rel||^2 ; segment max + count
__global__ void k_edge3(const float* __restrict__ ent, const int* __restrict__ eidx,
                        const int* __restrict__ etype, const float* __restrict__ exp1,
                        const float* __restrict__ sum1, const float* __restrict__ rr,
                        float* __restrict__ score_t, unsigned* __restrict__ max2,
                        float* __restrict__ cnt) {
  int e = blockIdx.x * blockDim.x + threadIdx.x;
  if (e >= N_EDGES) return;
  int h = eidx[e];
  int t = eidx[N_EDGES + e];
  int r = etype[e] - 1;
  float ra = exp1[e] / sum1[h];
  const float4* hr = (const float4*)(ent + (long)h * CH);
  const float4* tr = (const float4*)(ent + (long)t * CH);
  float s = 0.0f;
#pragma unroll
  for (int i = 0; i < 16; ++i) {
    float4 a = hr[i], b = tr[i];
    s += a.x * b.x + a.y * b.y + a.z * b.z + a.w * b.w;
  }
  float st = s + ra * ra * rr[r];
  score_t[e] = st;
  atomicMax(max2 + h, f2ord(st));
  atomicAdd(cnt + h, 1.0f);
}

// pass 5: scatter t_emb * weight[rel] * kg_mask into entity rows of out (16 threads/edge)
__global__ void k_edge5(const float* __restrict__ ent, const int* __restrict__ eidx,
                        const int* __restrict__ etype, const float* __restrict__ weight,
                        const float* __restrict__ exp2, const float* __restrict__ sum2,
                        float* __restrict__ out) {
  int gid = blockIdx.x * blockDim.x + threadIdx.x;
  int e = gid >> 4;
  if (e >= N_EDGES) return;
  int q4 = (gid & 15) * 4;
  int h = eidx[e];
  int t = eidx[N_EDGES + e];
  int r = etype[e] - 1;
  float mask = exp2[e] / sum2[h];
  float4 tv = *(const float4*)(ent + (long)t * CH + q4);
  float4 wv = *(const float4*)(weight + r * CH + q4);
  float* base = out + (long)h * CH + q4;
  atomicAdd(base + 0, tv.x * wv.x * mask);
  atomicAdd(base + 1, tv.y * wv.y * mask);
  atomicAdd(base + 2, tv.z * wv.z * mask);
  atomicAdd(base + 3, tv.w * wv.w * mask);
}

__global__ void k_norm(float* __restrict__ out, const float* __restrict__ cnt) {
  int i = blockIdx.x * blockDim.x + threadIdx.x;
  if (i >= N_ENT * CH) return;
  out[i] /= fmaxf(cnt[i >> 6], 1.0f);
}

// user_agg scatter: out[N_ENT+row] += ent[col] * val   (16 threads/nz)
__global__ void k_uscat(const float* __restrict__ ent, const int* __restrict__ uir,
                        const int* __restrict__ uic, const float* __restrict__ uiv,
                        float* __restrict__ out) {
  int gid = blockIdx.x * blockDim.x + threadIdx.x;
  int e = gid >> 4;
  if (e >= NNZ_) return;
  int q4 = (gid & 15) * 4;
  int row = uir[e];
  int col = uic[e];
  float v = uiv[e];
  float4 ev = *(const float4*)(ent + (long)col * CH + q4);
  float* base = out + ((long)(N_ENT + row)) * CH + q4;
  atomicAdd(base + 0, ev.x * v);
  atomicAdd(base + 1, ev.y * v);
  atomicAdd(base + 2, ev.z * v);
  atomicAdd(base + 3, ev.w * v);
}

// WMMA user finalize: S = softmax(U Lt) via 16 wmma K-steps; mult = S @ disen via 4 wmma;
// out_user = user_agg * (1 + mult), in place on out rows [N_ENT, N_ENT+N_USERS).
// One wave = 16 users. f32 16x16x4 layout: A lanes0-15 hold K{0,1}, lanes16-31 K{2,3};
// B VGPR0/1 hold K{half*2, half*2+1}, N=lane&15; C/D vgpr v: M = v + (lane>=16)*8, N=lane&15.
__global__ void k_user_final(const float* __restrict__ user, const float* __restrict__ latent,
                             const float* __restrict__ ws, float* __restrict__ out) {
  __shared__ float sP[8 * 64];
  const int lane = threadIdx.x & 31;
  const int warp = threadIdx.x >> 5;
  const int half = lane >> 4;
  const int l16 = lane & 15;
  int wgid = blockIdx.x * (blockDim.x >> 5) + warp;
  int u0 = wgid * 16;
  if (u0 >= N_USERS) return;  // wave-uniform: EXEC stays all-ones for active waves
  const float* dis = ws + OFF_DIS;

  // Branchless zero-padding for factors 4..15: clamp the row address, scale by 0/1.
  // Avoids s_and_saveexec / cbranch between consecutive WMMAs.
  const int   nclamp = (l16 < NF) ? l16 : 0;
  const float bsel   = (l16 < NF) ? 1.0f : 0.0f;
  const float* brow  = latent + nclamp * CH;

  // GEMM1: scores = U(16x64) @ latent^T(64x4, zero-padded to 16 cols)
  v8f acc;
#pragma unroll
  for (int i = 0; i < 8; ++i) acc[i] = 0.0f;
#pragma unroll
  for (int kk = 0; kk < 16; ++kk) {
    int k0 = kk * 4 + half * 2;
    float2 av = *(const float2*)(user + (long)(u0 + l16) * CH + k0);
    float2 bv = *(const float2*)(brow + k0);
    v2f a; a.x = av.x;        a.y = av.y;
    v2f b; b.x = bv.x * bsel; b.y = bv.y * bsel;
    acc = __builtin_amdgcn_wmma_f32_16x16x4_f32(false, a, false, b, (short)0, acc, false, false);
  }

  // softmax across factors (quad-lane reduction), stash probs to LDS in A-matrix order
  float* myP = sP + warp * 64;
#pragma unroll
  for (int v = 0; v < 8; ++v) {
    float s = acc[v];  // score for user row (v + half*8), factor n = l16 (valid n<4; padded n give 0)
    float m = s;
    m = fmaxf(m, __shfl_xor(m, 1, 32));
    m = fmaxf(m, __shfl_xor(m, 2, 32));
    float e = __expf(s - m);
    float d = e;
    d += __shfl_xor(d, 1, 32);
    d += __shfl_xor(d, 2, 32);
    if (l16 < NF) myP[(v + half * 8) * NF + l16] = e / d;
  }
  asm volatile("s_wait_dscnt 0" ::: "memory");

  // GEMM2: mult(16x64) = S(16x4) @ disen(4x64), one K-step per 16-col tile
  v2f a2;
  a2.x = myP[l16 * NF + half * 2 + 0];
  a2.y = myP[l16 * NF + half * 2 + 1];
#pragma unroll
  for (int t = 0; t < 4; ++t) {
    int n0 = t * 16;
    v2f b2;
    b2.x = dis[(half * 2 + 0) * CH + n0 + l16];
    b2.y = dis[(half * 2 + 1) * CH + n0 + l16];
    v8f mm;
#pragma unroll
    for (int i = 0; i < 8; ++i) mm[i] = 0.0f;
    mm = __builtin_amdgcn_wmma_f32_16x16x4_f32(false, a2, false, b2, (short)0, mm, false, false);
#pragma unroll
    for (int v = 0; v < 8; ++v) {
      int u = u0 + v + half * 8;
      long o = ((long)(N_ENT + u)) * CH + n0 + l16;
      out[o] = out[o] * (1.0f + mm[v]);
    }
  }
}

extern "C" void kernel_launch(void* const* d_in, const int* in_sizes, int n_in,
                              void* d_out, int out_size, void* d_ws, size_t ws_size,
                              hipStream_t stream) {
  const float* ent    = (const float*)d_in[0];
  const float* user   = (const float*)d_in[1];
  const float* latent = (const float*)d_in[2];
  const int*   eidx   = (const int*)d_in[3];
  const int*   etype  = (const int*)d_in[4];
  const int*   uir    = (const int*)d_in[5];
  const int*   uic    = (const int*)d_in[6];
  const float* uiv    = (const float*)d_in[7];
  const float* weight = (const float*)d_in[8];
  const float* datt   = (const float*)d_in[9];
  const float* W      = (const float*)d_in[10];
  float* out = (float*)d_out;
  float* ws  = (float*)d_ws;

  // zero: whole output (used as atomic accumulator) + softmax state (max-as-ord-u32 init 0 == -inf)
  k_zero<<<4096, 256, 0, stream>>>(out, (long)(N_ENT + N_USERS) * CH);
  k_zero<<<2048, 256, 0, stream>>>(ws + OFF_MAX1, (long)5 * N_ENT);

  k_pre<<<1, 256, 0, stream>>>(weight, datt, W, ws);

  int eb = (N_EDGES + 255) / 256;
  k_edge1<<<eb, 256, 0, stream>>>(ent, eidx, etype, ws + OFF_V, ws + OFF_SR,
                                  (unsigned*)(ws + OFF_MAX1));
  k_exp_sum<<<eb, 256, 0, stream>>>(eidx, ws + OFF_SR,
                                    (const unsigned*)(ws + OFF_MAX1), ws + OFF_SUM1);
  k_edge3<<<eb, 256, 0, stream>>>(ent, eidx, etype, ws + OFF_SR, ws + OFF_SUM1,
                                  ws + OFF_RR, ws + OFF_ST,
                                  (unsigned*)(ws + OFF_MAX2), ws + OFF_CNT);
  k_exp_sum<<<eb, 256, 0, stream>>>(eidx, ws + OFF_ST,
                                    (const unsigned*)(ws + OFF_MAX2), ws + OFF_SUM2);
  int eb16 = (N_EDGES * 16 + 255) / 256;
  k_edge5<<<eb16, 256, 0, stream>>>(ent, eidx, etype, weight, ws + OFF_ST,
                                    ws + OFF_SUM2, out);
  int nb = (N_ENT * CH + 255) / 256;
  k_norm<<<nb, 256, 0, stream>>>(out, ws + OFF_CNT);

  int ub16 = (NNZ_ * 16 + 255) / 256;
  k_uscat<<<ub16, 256, 0, stream>>>(ent, uir, uic, uiv, out);
  int fb = ((N_USERS / 16) + 7) / 8;  // 8 waves/block, 16 users/wave
  k_user_final<<<fb, 256, 0, stream>>>(user, latent, ws, out);
}